// KNN_5454608466219
// MI455X (gfx1250) — compile-verified
//
#include <hip/hip_runtime.h>

typedef __attribute__((ext_vector_type(2))) float v2f;
typedef __attribute__((ext_vector_type(8))) float v8f;

#define WAVES_PER_BLOCK 8
#define KSEL 16
#define COLPAD 20   // floats per column slot: rows 0-7 at +0, rows 8-15 at +12 (16B-aligned b128 stores, conflict-free scan)

// Prologue: Bprep[n] = (-2x, -2y, -2z, |x|^2)  -- 1 MB, lives in L2 for the whole run.
__global__ void knn_prep_kernel(const float* __restrict__ pts, float4* __restrict__ prep, int N) {
    int n = blockIdx.x * blockDim.x + threadIdx.x;
    if (n < N) {
        float x = pts[3 * n + 0], y = pts[3 * n + 1], z = pts[3 * n + 2];
        float4 o;
        o.x = -2.0f * x; o.y = -2.0f * y; o.z = -2.0f * z;
        o.w = x * x + y * y + z * z;
        prep[n] = o;
    }
}

template <bool USE_PREP>
__device__ __forceinline__ v2f load_b(const float* __restrict__ pts,
                                      const float2* __restrict__ prep2,
                                      int n0, int colId, int hi) {
    v2f b;
    if (USE_PREP) {
        float2 bp = prep2[((n0 + colId) << 1) + hi];
        b[0] = bp.x; b[1] = bp.y;
    } else {
        int n = n0 + colId;
        float x = pts[3 * n + 0], y = pts[3 * n + 1], z = pts[3 * n + 2];
        float nrm = x * x + y * y + z * z;
        b[0] = hi ? (-2.0f * z) : (-2.0f * x);
        b[1] = hi ? nrm         : (-2.0f * y);
    }
    return b;
}

template <bool USE_PREP>
__global__ __launch_bounds__(256) void knn_main_kernel(
    const float* __restrict__ pts,
    const float2* __restrict__ prep2,   // prep viewed as pairs: [n*2+0]=(-2x,-2y), [n*2+1]=(-2z,|x|^2)
    float* __restrict__ out, int N) {

    __shared__ float tile[WAVES_PER_BLOCK][16 * COLPAD];
    __shared__ float mv[WAVES_PER_BLOCK][16][KSEL];
    __shared__ int   mi[WAVES_PER_BLOCK][16][KSEL];

    const int lane  = threadIdx.x & 31;
    const int wave  = threadIdx.x >> 5;
    const int colId = lane & 15;     // my column inside the 16-wide tile; also my owned row
    const int hi    = lane >> 4;     // half-wave select (K-split for A/B, col-range split for scan)

    const int qbase = blockIdx.x * (WAVES_PER_BLOCK * 16) + wave * 16;
    if (qbase >= N) return;

    // ---- A operand: 16x4 fp32, homogeneous coords (qx,qy,qz,1).
    // Layout: lanes 0-15 hold K=0(v0),K=1(v1); lanes 16-31 hold K=2(v0),K=3(v1).
    v2f a;
    {
        int m = qbase + colId;
        float x = pts[3 * m + 0], y = pts[3 * m + 1], z = pts[3 * m + 2];
        a[0] = hi ? z : x;
        a[1] = hi ? 1.0f : y;
    }

    // ---- register-resident sorted top-16 (ascending; [15] = current worst)
    float topv[KSEL]; int topi[KSEL];
#pragma unroll
    for (int j = 0; j < KSEL; ++j) { topv[j] = 3.4e38f; topi[j] = 0; }

    const int row   = colId;                    // row this lane selects for
    const int foff  = row + ((row >> 3) << 2);  // LDS row offset (rows 8-15 shifted +4)
    const int cbase = hi * 8;                   // this lane scans cols cbase..cbase+7
    float* tw = tile[wave];
    float* st = &tw[colId * COLPAD + (hi ? 12 : 0)];

    const int nTiles = N >> 4;

    // Software pipeline: B tile for iteration t is loaded during iteration t-1,
    // so the s_wait_loadcnt before each WMMA covers a load issued a full loop
    // body (LDS scan + insert work) earlier -> L2 latency off the critical path.
    v2f b = load_b<USE_PREP>(pts, prep2, 0, colId, hi);

    for (int t = 0; t < nTiles; ++t) {
        const int n0 = t << 4;

        // ---- D = A x B : 16x16 tile of (|x|^2 - 2 q.x); |q|^2 dropped (row-constant)
        v8f c = {};
        c = __builtin_amdgcn_wmma_f32_16x16x4_f32(false, a, false, b, (short)0, c, false, false);

        // ---- issue next tile's B load now (consumed next iteration)
        {
            int tn = t + 1; tn = (tn < nTiles) ? tn : t;   // clamp, branch-free
            b = load_b<USE_PREP>(pts, prep2, tn << 4, colId, hi);
        }

        // ---- stage tile to LDS: lane holds rows 8hi..8hi+7 of column colId
        *(float4*)(st + 0) = make_float4(c[0], c[1], c[2], c[3]);
        *(float4*)(st + 4) = make_float4(c[4], c[5], c[6], c[7]);
        __builtin_amdgcn_wave_barrier();   // same-wave LDS is in-order; just block compiler reordering

        // ---- scan my 8 columns of my row against current worst
        float vv[8];
        unsigned m = 0;
        float worst = topv[KSEL - 1];
#pragma unroll
        for (int i = 0; i < 8; ++i) {
            vv[i] = tw[(cbase + i) * COLPAD + foff];
            m |= (vv[i] < worst) ? (1u << i) : 0u;
        }

        if (m) {  // rare after warm-up: ~133 inserts per row over 4096 tiles
#pragma unroll
            for (int i = 0; i < 8; ++i) {
                if (m & (1u << i)) {
                    float v = vv[i]; int id = n0 + cbase + i;
#pragma unroll
                    for (int j = 0; j < KSEL; ++j) {   // unrolled sorted insert, stays in VGPRs
                        bool sw = v < topv[j];
                        float tv = topv[j]; int ti = topi[j];
                        topv[j] = sw ? v : tv;  topi[j] = sw ? id : ti;
                        v = sw ? tv : v;        id = sw ? ti : id;
                    }
                }
            }
        }
        __builtin_amdgcn_wave_barrier();
    }

    // ---- merge the two half-lists per row (lane r owns cols==0..7 mod 16, lane r+16 owns 8..15)
    if (hi) {
#pragma unroll
        for (int j = 0; j < KSEL; ++j) { mv[wave][row][j] = topv[j]; mi[wave][row][j] = topi[j]; }
    }
    __syncthreads();
    if (!hi) {
#pragma unroll
        for (int j = 0; j < KSEL; ++j) {
            float v = mv[wave][row][j]; int id = mi[wave][row][j];
#pragma unroll
            for (int jj = 0; jj < KSEL; ++jj) {
                bool sw = v < topv[jj];
                float tv = topv[jj]; int ti = topi[jj];
                topv[jj] = sw ? v : tv;  topi[jj] = sw ? id : ti;
                v = sw ? tv : v;         id = sw ? ti : id;
            }
        }
        // ---- emit indices as float, 16 contiguous per query -> 4x global_store_b128
        float* orow = out + (size_t)(qbase + row) * KSEL;
#pragma unroll
        for (int j = 0; j < KSEL; j += 4) {
            *(float4*)(orow + j) = make_float4((float)topi[j + 0], (float)topi[j + 1],
                                               (float)topi[j + 2], (float)topi[j + 3]);
        }
    }
}

extern "C" void kernel_launch(void* const* d_in, const int* in_sizes, int n_in,
                              void* d_out, int out_size, void* d_ws, size_t ws_size,
                              hipStream_t stream) {
    const float* pts = (const float*)d_in[0];
    const int N = in_sizes[0] / 3;          // (N, 3) float32; k is fixed at 16 per reference
    float* out = (float*)d_out;

    const int blocks = (N + WAVES_PER_BLOCK * 16 - 1) / (WAVES_PER_BLOCK * 16);
    const bool use_prep = ws_size >= (size_t)N * sizeof(float4);

    if (use_prep) {
        knn_prep_kernel<<<(N + 255) / 256, 256, 0, stream>>>(pts, (float4*)d_ws, N);
        knn_main_kernel<true><<<blocks, 256, 0, stream>>>(pts, (const float2*)d_ws, out, N);
    } else {
        knn_main_kernel<false><<<blocks, 256, 0, stream>>>(pts, nullptr, out, N);
    }
}